// ChamferDistanceSqrt_39530878992590
// MI455X (gfx1250) — compile-verified
//
#include <hip/hip_runtime.h>

typedef __attribute__((ext_vector_type(2))) float v2f;
typedef __attribute__((ext_vector_type(8))) float v8f;

__global__ void chamfer_zero_kernel(float* out, int n) {
    int i = blockIdx.x * blockDim.x + threadIdx.x;
    if (i < n) out[i] = 0.0f;
}

// One wave32 owns a 16-row tile of X for one batch and scans all M columns of Y
// in 16-wide tiles using V_WMMA_F32_16X16X4_F32.
// The squared distance is folded into a single K=4 GEMM (C = 0):
//   A (16x4) : row m = [-2x0, -2x1, -2x2, 1]
//              lanes 0-15  hold (K0,K1) = (-2x0, -2x1) for row = lane
//              lanes 16-31 hold (K2,K3) = (-2x2,  1.0) for row = lane-16
//   B (4x16) : col n = [y0, y1, y2, ||y||^2]
//              lanes 0-15  hold (K0,K1) = (y0, y1)       for col = lane
//              lanes 16-31 hold (K2,K3) = (y2, ||y||^2)  for col = lane-16
// => D[m][n] = ||y_n||^2 - 2 x_m . y_n.  Running min over column tiles; the
// row-uniform ||x_m||^2 is added after the min reduction (commutes with min_n).
// All loads are unconditional (every lane reads its point's 3 floats), so the
// inner loop is branch-free: fragment elements are lane-half selects.
__global__ __launch_bounds__(256) void chamfer_min_kernel(
    const float* __restrict__ X, const float* __restrict__ Y,
    float* __restrict__ out, int tilesPerBatch, int N, int M, float scale) {
    const int lane = threadIdx.x & 31;
    const int gw   = (blockIdx.x * blockDim.x + threadIdx.x) >> 5;
    const int b    = gw / tilesPerBatch;
    const int rt   = gw - b * tilesPerBatch;
    const bool hi  = (lane >= 16);

    const float* __restrict__ Xb = X + (size_t)b * N * 3;
    const float* __restrict__ Yb = Y + (size_t)b * M * 3;

    // ---- A fragment + row norm (branch-free) ----
    const int r = rt * 16 + (lane & 15);
    const float x0 = Xb[3 * r + 0];
    const float x1 = Xb[3 * r + 1];
    const float x2 = Xb[3 * r + 2];
    const float xn = x0 * x0 + x1 * x1 + x2 * x2;   // ||x_(lane&15)||^2, local

    v2f afrag;
    afrag.x = hi ? (-2.0f * x2) : (-2.0f * x0);
    afrag.y = hi ? 1.0f         : (-2.0f * x1);

    const v8f czero = {};   // C = 0

    v8f rmin;
#pragma unroll
    for (int v = 0; v < 8; ++v) rmin[v] = 3.402823466e+38f;

#pragma unroll 4
    for (int c = 0; c < M; c += 16) {
        const int p = c + (lane & 15);
        const float y0 = Yb[3 * p + 0];
        const float y1 = Yb[3 * p + 1];
        const float y2 = Yb[3 * p + 2];
        __builtin_prefetch(Yb + 3 * (p + 16), 0, 3);   // near-scope speculative prefetch
        const float yn = y0 * y0 + y1 * y1 + y2 * y2;  // ||y_(c + lane&15)||^2, local

        v2f bfrag;
        bfrag.x = hi ? y2 : y0;
        bfrag.y = hi ? yn : y1;

        // D = A x B + 0 = ||y_n||^2 - 2 x_m . y_n
        v8f d = __builtin_amdgcn_wmma_f32_16x16x4_f32(
            /*neg_a=*/false, afrag, /*neg_b=*/false, bfrag,
            /*c_mod=*/(short)0, czero, /*reuse_a=*/false, /*reuse_b=*/false);

#pragma unroll
        for (int v = 0; v < 8; ++v) rmin[v] = fminf(rmin[v], d[v]);
    }

    // Butterfly min within each 16-lane half (lanes 0-15: rows v; 16-31: rows v+8).
#pragma unroll
    for (int m = 8; m >= 1; m >>= 1) {
#pragma unroll
        for (int v = 0; v < 8; ++v)
            rmin[v] = fminf(rmin[v], __shfl_xor(rmin[v], m, 32));
    }

    // Add per-row ||x||^2, sqrt, and sum the 16 row distances of this tile.
    float sum = 0.0f;
#pragma unroll
    for (int v = 0; v < 8; ++v) {
        const int src  = hi ? (v + 8) : v;
        const float xv = __shfl(xn, src, 32);   // ||x_row||^2 for this half's row
        sum += sqrtf(fmaxf(xv + rmin[v], 0.0f));
    }
    // lanes 0 (rows 0-7) and 16 (rows 8-15) contribute the tile's partial mean
    if ((lane & 15) == 0) atomicAdd(out, sum * scale);
}

extern "C" void kernel_launch(void* const* d_in, const int* in_sizes, int n_in,
                              void* d_out, int out_size, void* d_ws, size_t ws_size,
                              hipStream_t stream) {
    const float* xyz1 = (const float*)d_in[0];
    const float* xyz2 = (const float*)d_in[1];
    float* out = (float*)d_out;

    const int B = 4;
    const int N = in_sizes[0] / (B * 3);   // 8192
    const int M = in_sizes[1] / (B * 3);   // 8192

    chamfer_zero_kernel<<<1, 64, 0, stream>>>(out, out_size);

    // Pass 1: dist1 — for each x, min over y.
    {
        const int tilesPerBatch = N / 16;
        const int waves = B * tilesPerBatch;            // 2048 waves
        const int threads = waves * 32;
        dim3 block(256);
        dim3 grid((threads + 255) / 256);
        chamfer_min_kernel<<<grid, block, 0, stream>>>(
            xyz1, xyz2, out, tilesPerBatch, N, M, 1.0f / (float)(B * N));
    }
    // Pass 2: dist2 — for each y, min over x (same kernel, operands swapped).
    {
        const int tilesPerBatch = M / 16;
        const int waves = B * tilesPerBatch;
        const int threads = waves * 32;
        dim3 block(256);
        dim3 grid((threads + 255) / 256);
        chamfer_min_kernel<<<grid, block, 0, stream>>>(
            xyz2, xyz1, out, tilesPerBatch, M, N, 1.0f / (float)(B * M));
    }
}